// KNNQueryAndGroup_48215302865113
// MI455X (gfx1250) — compile-verified
//
#include <hip/hip_runtime.h>
#include <stdint.h>

#define B_       4
#define N_       16384
#define NPOINT_  4096
#define C_       64
#define NSAMPLE_ 32
#define CH_      (3 + C_)

#define WAVES_PER_WG 4
#define QPW          16                 // queries per wave (WMMA M dim)
#define THREADS      (WAVES_PER_WG * 32)
#define CAP          64                 // collect-buffer capacity per query

typedef float v2f __attribute__((ext_vector_type(2)));
typedef float v8f __attribute__((ext_vector_type(8)));

// monotone float -> uint mapping (handles negatives)
__device__ __forceinline__ uint32_t fkey(float f) {
  uint32_t u = __float_as_uint(f);
  return u ^ ((uint32_t)((int32_t)u >> 31) | 0x80000000u);
}

__device__ __forceinline__ uint32_t shfl_up_u32(uint32_t v, int d) {
  return (uint32_t)__shfl_up((int)v, d, 32);
}
__device__ __forceinline__ unsigned long long shfl_xor_u64(unsigned long long v, int m) {
  uint32_t lo = (uint32_t)__shfl_xor((int)(uint32_t)v, m, 32);
  uint32_t hi = (uint32_t)__shfl_xor((int)(uint32_t)(v >> 32), m, 32);
  return ((unsigned long long)hi << 32) | (unsigned long long)lo;
}

__global__ __launch_bounds__(THREADS)
void knn_group_wmma(const float* __restrict__ xyz,
                    const float* __restrict__ new_xyz,
                    const float* __restrict__ feat,
                    float* __restrict__ out)
{
  // 32KB region, time-shared: u32 view = 256-bin u16 histograms (128 words/query),
  // u64 view = per-query (key,idx) collect buffers. Phases split by __syncthreads().
  __shared__ unsigned long long s_u64[WAVES_PER_WG * QPW * CAP];
  __shared__ uint32_t s_idx[WAVES_PER_WG * QPW * NSAMPLE_];
  __shared__ uint32_t s_b1 [WAVES_PER_WG * QPW];
  __shared__ uint32_t s_cb1[WAVES_PER_WG * QPW];
  __shared__ uint32_t s_P  [WAVES_PER_WG * QPW];
  __shared__ uint32_t s_cnt[WAVES_PER_WG * QPW];

  uint32_t* hist = (uint32_t*)s_u64;

  const int lane = (int)(threadIdx.x & 31u);
  const int wave = (int)(threadIdx.x >> 5);
  const int hi16 = lane >> 4;   // 0: lanes 0-15, 1: lanes 16-31
  const int l16  = lane & 15;
  const int wq   = wave * QPW;

  const int flatQ0 = ((int)blockIdx.x * WAVES_PER_WG + wave) * QPW;
  const int b  = flatQ0 / NPOINT_;          // 64 queries/WG divides NPOINT -> uniform batch
  const int q0 = flatQ0 - b * NPOINT_;

  const float* __restrict__ xyzB = xyz + (size_t)b * N_ * 3;

  // A fragment (16x4 f32): row m = l16 holds (-2qx, -2qy, -2qz, 1).
  // Lanes 0-15 carry K=0,1 ; lanes 16-31 carry K=2,3 (ISA 7.12.2).
  const float* qp = new_xyz + ((size_t)b * NPOINT_ + (size_t)(q0 + l16)) * 3;
  const float qx = qp[0], qy = qp[1], qz = qp[2];
  v2f afrag;
  afrag.x = hi16 ? (-2.0f * qz) : (-2.0f * qx);
  afrag.y = hi16 ? 1.0f         : (-2.0f * qy);

  // One 16(query) x 16(candidate) tile of s = ||p||^2 - 2 q.p  (order-equivalent to dist)
  auto dist_tile = [&](int base) -> v8f {
    const float* p = xyzB + (size_t)(base + l16) * 3;
    const float px = p[0], py = p[1], pz = p[2];
    const float sq = px * px + py * py + pz * pz;
    v2f bfrag;                       // B col n = l16: (px,py,pz,sq); K-halves per lane-half
    bfrag.x = hi16 ? pz : px;
    bfrag.y = hi16 ? sq : py;
    v8f acc = {0.f, 0.f, 0.f, 0.f, 0.f, 0.f, 0.f, 0.f};
    return __builtin_amdgcn_wmma_f32_16x16x4_f32(false, afrag, false, bfrag,
                                                 (short)0, acc, false, false);
  };

  // scan one query's 256-bin histogram; cumulative seeded by 'seed';
  // single crossing lane writes bin and strictly-below count.
  auto scan_hist = [&](int q, uint32_t seed, uint32_t* outBin, uint32_t* outBelow) {
    uint32_t cnt8[8];
    #pragma unroll
    for (int w = 0; w < 4; ++w) {
      uint32_t word = hist[(wq + q) * 128 + 4 * lane + w];
      cnt8[2 * w]     = word & 0xFFFFu;
      cnt8[2 * w + 1] = word >> 16;
    }
    uint32_t lsum = 0;
    #pragma unroll
    for (int i = 0; i < 8; ++i) lsum += cnt8[i];
    uint32_t incl = lsum;
    #pragma unroll
    for (int off = 1; off < 32; off <<= 1) {
      uint32_t t = shfl_up_u32(incl, off);
      if (lane >= off) incl += t;
    }
    uint32_t excl = seed + incl - lsum;
    uint32_t inc2 = seed + incl;
    if (excl < NSAMPLE_ && inc2 >= NSAMPLE_) {
      uint32_t run = excl;
      #pragma unroll
      for (int i = 0; i < 8; ++i) {
        if (run < NSAMPLE_ && run + cnt8[i] >= NSAMPLE_) {
          *outBin   = (uint32_t)(8 * lane + i);
          *outBelow = run;
        }
        run += cnt8[i];
      }
    }
  };

  // ---- phase 0: clear histograms ----
  for (int i = (int)threadIdx.x; i < WAVES_PER_WG * QPW * 128; i += THREADS) hist[i] = 0u;
  __syncthreads();

  // ---- pass 1: histogram key[31:24] ----
  for (int base = 0; base < N_; base += 16) {
    v8f d = dist_tile(base);
    #pragma unroll
    for (int r = 0; r < 8; ++r) {
      const int qm = r + 8 * hi16;                 // D layout: lanes<16 -> M=r, else M=r+8
      const uint32_t key = fkey(d[r]);
      const uint32_t bin = key >> 24;
      atomicAdd(&hist[(wq + qm) * 128 + (bin >> 1)], 1u << ((bin & 1u) * 16));
    }
  }
  __syncthreads();

  for (int q = 0; q < QPW; ++q) scan_hist(q, 0u, &s_b1[wq + q], &s_cb1[wq + q]);
  __syncthreads();

  // ---- clear histograms again ----
  for (int i = (int)threadIdx.x; i < WAVES_PER_WG * QPW * 128; i += THREADS) hist[i] = 0u;
  __syncthreads();

  // ---- pass 2: histogram key[23:16] within level-1 bin ----
  uint32_t b1v[8];
  #pragma unroll
  for (int r = 0; r < 8; ++r) b1v[r] = s_b1[wq + (r + 8 * hi16)];
  for (int base = 0; base < N_; base += 16) {
    v8f d = dist_tile(base);
    #pragma unroll
    for (int r = 0; r < 8; ++r) {
      const int qm = r + 8 * hi16;
      const uint32_t key = fkey(d[r]);
      if ((key >> 24) == b1v[r]) {
        const uint32_t bin = (key >> 16) & 0xFFu;
        atomicAdd(&hist[(wq + qm) * 128 + (bin >> 1)], 1u << ((bin & 1u) * 16));
      }
    }
  }
  __syncthreads();

  for (int q = 0; q < QPW; ++q) scan_hist(q, s_cb1[wq + q], &s_P[wq + q], &s_cnt[wq + q]);
  if (lane < QPW) s_P[wq + lane] = (s_b1[wq + lane] << 8) | (s_P[wq + lane] & 0xFFu);
  __syncthreads();

  // ---- init collect buffers (alias over hist) + counters ----
  for (int i = (int)threadIdx.x; i < WAVES_PER_WG * QPW * CAP; i += THREADS) s_u64[i] = ~0ull;
  if (threadIdx.x < WAVES_PER_WG * QPW) s_cnt[threadIdx.x] = 0u;
  __syncthreads();

  // ---- pass 3: collect every candidate with key[31:16] <= P (superset of top-32) ----
  uint32_t Pv[8];
  #pragma unroll
  for (int r = 0; r < 8; ++r) Pv[r] = s_P[wq + (r + 8 * hi16)];
  for (int base = 0; base < N_; base += 16) {
    v8f d = dist_tile(base);
    #pragma unroll
    for (int r = 0; r < 8; ++r) {
      const int qm = r + 8 * hi16;
      const uint32_t key = fkey(d[r]);
      if ((key >> 16) <= Pv[r]) {
        uint32_t pos = atomicAdd(&s_cnt[wq + qm], 1u);
        if (pos < CAP)
          s_u64[(wq + qm) * CAP + pos] =
              ((unsigned long long)key << 32) | (unsigned long long)(uint32_t)(base + l16);
      }
    }
  }
  __syncthreads();

  // ---- extract 32 smallest (key, idx) per query, ascending (matches stable top_k) ----
  for (int q = 0; q < QPW; ++q) {
    unsigned long long* bq = &s_u64[(wq + q) * CAP];
    for (int round = 0; round < NSAMPLE_; ++round) {
      unsigned long long v0 = bq[lane];
      unsigned long long v1 = bq[lane + 32];
      unsigned long long m = v0 < v1 ? v0 : v1;
      #pragma unroll
      for (int off = 16; off > 0; off >>= 1) {
        unsigned long long o = shfl_xor_u64(m, off);
        m = o < m ? o : m;
      }
      if (v0 == m)      bq[lane]      = ~0ull;   // idx part unique -> single winner slot
      else if (v1 == m) bq[lane + 32] = ~0ull;
      if (lane == 0) s_idx[(wq + q) * NSAMPLE_ + round] = (uint32_t)m;
    }
  }
  __syncthreads();

  // ---- gather + write: out (B, 3+C, NPOINT, NSAMPLE), lane = sample ----
  const size_t chStride = (size_t)NPOINT_ * NSAMPLE_;
  for (int q = 0; q < QPW; ++q) {
    const int qq = q0 + q;
    const uint32_t j = s_idx[(wq + q) * NSAMPLE_ + lane];
    const float* p = xyzB + (size_t)j * 3;
    const float px = p[0], py = p[1], pz = p[2];
    const float* qc = new_xyz + ((size_t)b * NPOINT_ + (size_t)qq) * 3;
    float* o = out + (((size_t)b * CH_) * NPOINT_ + (size_t)qq) * NSAMPLE_ + lane;
    o[0]            = px - qc[0];
    o[chStride]     = py - qc[1];
    o[2 * chStride] = pz - qc[2];
    const float* f = feat + ((size_t)b * C_) * N_ + j;
    float* o2 = o + 3 * chStride;
    #pragma unroll 8
    for (int ch = 0; ch < C_; ++ch)
      o2[(size_t)ch * chStride] = f[(size_t)ch * N_];
  }
}

extern "C" void kernel_launch(void* const* d_in, const int* in_sizes, int n_in,
                              void* d_out, int out_size, void* d_ws, size_t ws_size,
                              hipStream_t stream) {
  (void)in_sizes; (void)n_in; (void)d_ws; (void)ws_size; (void)out_size;
  const float* xyz     = (const float*)d_in[0];
  const float* new_xyz = (const float*)d_in[1];
  const float* feat    = (const float*)d_in[2];
  float* out           = (float*)d_out;
  dim3 grid((B_ * NPOINT_) / (WAVES_PER_WG * QPW));   // 256 WGs x 128 threads
  knn_group_wmma<<<grid, THREADS, 0, stream>>>(xyz, new_xyz, feat, out);
}